// PQVQVAE_58712202936472
// MI455X (gfx1250) — compile-verified
//
#include <hip/hip_runtime.h>
#include <math.h>

typedef __attribute__((ext_vector_type(2))) float v2f;
typedef __attribute__((ext_vector_type(8))) float v8f;

__device__ __forceinline__ v8f wmma_f32(v2f a, v2f b, v8f c) {
    return __builtin_amdgcn_wmma_f32_16x16x4_f32(false, a, false, b, (short)0, c, false, false);
}

// CDNA5 async global->LDS copy (ASYNCcnt-tracked), per CDNA5 ISA.
// VDST = per-lane LDS byte address, VADDR = 64-bit global address.
__device__ __forceinline__ void async_copy_b128_to_lds(unsigned lds, const void* gaddr) {
    asm volatile("global_load_async_to_lds_b128 %0, %1, off" :: "v"(lds), "v"(gaddr) : "memory");
}
__device__ __forceinline__ void wait_asynccnt0() {
    asm volatile("s_wait_asynccnt 0x0" ::: "memory");
}

// ---------------------------------------------------------------------------
// Conv1d inner K-loop (implicit GEMM, V_WMMA_F32_16X16X4_F32).
// ---------------------------------------------------------------------------
template<int KW, bool RELU, bool EDGE, bool INTERIOR, int NT>
__device__ __forceinline__ void conv_kloop(const float* __restrict__ wrow,
                                           const float* __restrict__ xb,
                                           int Ktot, int Cin, int Tin, int co, int Cout,
                                           int koff, const int* tbase, v8f* acc)
{
    for (int kk = 0; kk < Ktot; kk += 4) {
        const int kb0 = kk + koff, kb1 = kb0 + 1;
        v2f a;
        if (EDGE) {
            int kc0 = kb0 < Ktot ? kb0 : Ktot - 1;
            int kc1 = kb1 < Ktot ? kb1 : Ktot - 1;
            float a0 = wrow[kc0], a1 = wrow[kc1];
            a.x = (co < Cout && kb0 < Ktot) ? a0 : 0.f;
            a.y = (co < Cout && kb1 < Ktot) ? a1 : 0.f;
        } else {
            a = *(const v2f*)(wrow + kb0);          // 8B-aligned b64 load
        }
        int ci0 = kb0 / KW; const int kw0 = kb0 - ci0 * KW;   // constant-divisor
        int ci1 = kb1 / KW; const int kw1 = kb1 - ci1 * KW;
        if (EDGE) { ci0 = ci0 < Cin ? ci0 : Cin - 1; ci1 = ci1 < Cin ? ci1 : Cin - 1; }
        const float* p0 = xb + (size_t)ci0 * Tin;
        const float* p1 = xb + (size_t)ci1 * Tin;
        #pragma unroll
        for (int q = 0; q < NT; ++q) {
            const int t0i = tbase[q] + kw0;
            const int t1i = tbase[q] + kw1;
            float v0, v1;
            if (INTERIOR) {
                v0 = p0[t0i];
                v1 = p1[t1i];
            } else {
                int c0 = t0i < 0 ? 0 : (t0i >= Tin ? Tin - 1 : t0i);
                int c1 = t1i < 0 ? 0 : (t1i >= Tin ? Tin - 1 : t1i);
                v0 = p0[c0]; v1 = p1[c1];
                v0 = (t0i == c0) ? v0 : 0.f;
                v1 = (t1i == c1) ? v1 : 0.f;
            }
            if (EDGE) { if (kb0 >= Ktot) v0 = 0.f; if (kb1 >= Ktot) v1 = 0.f; }
            if (RELU) { v0 = fmaxf(v0, 0.f); v1 = fmaxf(v1, 0.f); }
            v2f bf; bf.x = v0; bf.y = v1;
            acc[q] = wmma_f32(a, bf, acc[q]);
        }
    }
}

template<int KW, int STRIDE, int PAD, bool RELU, bool ADD, bool EDGE, int NT>
__global__ __launch_bounds__(128)
void conv1d_wmma_kernel(const float* __restrict__ x, const float* __restrict__ w,
                        const float* __restrict__ bias, const float* __restrict__ addend,
                        float* __restrict__ y,
                        int B, int Cin, int Tin, int Cout, int Tout)
{
    const int lane = threadIdx.x & 31;
    const int wave = blockIdx.x * (blockDim.x >> 5) + (threadIdx.x >> 5);
    const int coTiles = (Cout + 15) >> 4;
    const int tGroups = Tout / (16 * NT);
    const int perImg  = coTiles * tGroups;
    if (wave >= B * perImg) return;                 // wave-uniform exit
    const int b = wave / perImg;
    int rem = wave - b * perImg;
    const int coT = rem / tGroups;
    const int tG  = rem - coT * tGroups;

    const int m    = lane & 15;
    const int koff = (lane >> 4) << 1;
    const int Ktot = Cin * KW;
    const int co   = coT * 16 + m;
    const int coc  = EDGE ? (co < Cout ? co : Cout - 1) : co;
    const float* wrow = w + (size_t)coc * Ktot;
    const float* xb   = x + (size_t)b * Cin * Tin;
    const int t0 = tG * (16 * NT);

    int tbase[NT];
    #pragma unroll
    for (int q = 0; q < NT; ++q) tbase[q] = (t0 + q * 16 + m) * STRIDE - PAD;

    __builtin_prefetch(wrow, 0, 0);                                  // global_prefetch_b8
    __builtin_prefetch(xb + (tbase[0] < 0 ? 0 : tbase[0]), 0, 0);

    v8f acc[NT] = {};
    const bool interior = (t0 * STRIDE - PAD >= 0) &&
                          ((t0 + 16 * NT - 1) * STRIDE + (KW - 1) - PAD < Tin);
    if (interior)
        conv_kloop<KW, RELU, EDGE, true,  NT>(wrow, xb, Ktot, Cin, Tin, co, Cout, koff, tbase, acc);
    else
        conv_kloop<KW, RELU, EDGE, false, NT>(wrow, xb, Ktot, Cin, Tin, co, Cout, koff, tbase, acc);

    const int nC = lane & 15;
    const int mB = (lane >> 4) * 8;
    float bv[8];
    #pragma unroll
    for (int v = 0; v < 8; ++v) {
        int coO = coT * 16 + mB + v;
        bv[v] = (!EDGE || coO < Cout) ? bias[coO] : 0.f;
    }
    #pragma unroll
    for (int q = 0; q < NT; ++q) {
        #pragma unroll
        for (int v = 0; v < 8; ++v) {
            int coO = coT * 16 + mB + v;
            if (EDGE && coO >= Cout) continue;
            size_t off = ((size_t)b * Cout + coO) * Tout + (size_t)t0 + q * 16 + nC;
            float r = acc[q][v] + bv[v];
            if (ADD) r += addend[off];
            y[off] = r;
        }
    }
}

// ---------------------------------------------------------------------------
// ConvTranspose1d (K=4, stride=2, pad=1) per-parity implicit GEMM.
// ---------------------------------------------------------------------------
template<bool INTERIOR, int NT>
__device__ __forceinline__ void convt_kloop(const float* __restrict__ w,
                                            const float* __restrict__ xb,
                                            int Cout, int Tin, int co, int k0, int koff,
                                            int Ktot, const int* txA, v8f* acc)
{
    for (int kk = 0; kk < Ktot; kk += 4) {
        const int ci = (kk + koff) >> 1;
        const float4 wv = *(const float4*)(w + ((size_t)ci * Cout + co) * 4);
        v2f a;
        a.x = k0 ? wv.y : wv.x;                     // tap k0
        a.y = k0 ? wv.w : wv.z;                     // tap k0+2
        const float* xc = xb + (size_t)ci * Tin;
        #pragma unroll
        for (int q = 0; q < NT; ++q) {
            float v0, v1;
            if (INTERIOR) {
                v0 = xc[txA[q]];
                v1 = xc[txA[q] - 1];
            } else {
                int t0i = txA[q], t1i = txA[q] - 1;
                int c0 = t0i < 0 ? 0 : (t0i >= Tin ? Tin - 1 : t0i);
                int c1 = t1i < 0 ? 0 : (t1i >= Tin ? Tin - 1 : t1i);
                v0 = xc[c0]; v1 = xc[c1];
                v0 = (t0i == c0) ? v0 : 0.f;
                v1 = (t1i == c1) ? v1 : 0.f;
            }
            v2f bf; bf.x = v0; bf.y = v1;
            acc[q] = wmma_f32(a, bf, acc[q]);
        }
    }
}

template<int NT>
__global__ __launch_bounds__(128)
void convT1d_wmma_kernel(const float* __restrict__ x, const float* __restrict__ w,
                         const float* __restrict__ bias, float* __restrict__ y,
                         int B, int Cin, int Tin, int Cout)
{
    const int Tout = Tin * 2;
    const int lane = threadIdx.x & 31;
    const int wave = blockIdx.x * (blockDim.x >> 5) + (threadIdx.x >> 5);
    const int coTiles = Cout >> 4;
    const int pG = Tout / (32 * NT);
    const int perImg = coTiles * 2 * pG;
    if (wave >= B * perImg) return;
    int b = wave / perImg;
    int rem = wave - b * perImg;
    const int coT = rem / (2 * pG); rem -= coT * 2 * pG;
    const int par = rem / pG;
    const int tG  = rem - par * pG;

    const int m    = lane & 15;
    const int koff = (lane >> 4) << 1;
    const int k0   = (par + 1) & 1;
    const int co   = coT * 16 + m;
    const int Ktot = Cin * 2;
    const float* xb = x + (size_t)b * Cin * Tin;

    int txA[NT];
    #pragma unroll
    for (int q = 0; q < NT; ++q) {
        int tcol = ((tG * NT + q) * 16 + m) * 2 + par;
        txA[q] = (tcol + 1 - k0) >> 1;
    }
    __builtin_prefetch(w + ((size_t)(koff >> 1) * Cout + co) * 4, 0, 0);

    v8f acc[NT] = {};
    const bool interior = !((par == 0 && tG == 0) || (par == 1 && tG == pG - 1));
    if (interior) convt_kloop<true,  NT>(w, xb, Cout, Tin, co, k0, koff, Ktot, txA, acc);
    else          convt_kloop<false, NT>(w, xb, Cout, Tin, co, k0, koff, Ktot, txA, acc);

    const int nC = lane & 15;
    const int mB = (lane >> 4) * 8;
    #pragma unroll
    for (int v = 0; v < 8; ++v) {
        const int coO = coT * 16 + mB + v;
        const float bvv = bias[coO];
        #pragma unroll
        for (int q = 0; q < NT; ++q) {
            int tO = ((tG * NT + q) * 16 + nC) * 2 + par;
            y[((size_t)b * Cout + coO) * Tout + tO] = acc[q][v] + bvv;
        }
    }
}

// ---------------------------------------------------------------------------
// VQ argmin with async-LDS codebook staging.
// Block = 4 waves x 16 points. Each 16-code tile (4KB, contiguous in emb) is
// staged into LDS once per block via GLOBAL_LOAD_ASYNC_TO_LDS_B128 (double-
// buffered, overlapped with WMMA on the previous tile), then all 4 waves read
// B-fragments from LDS. ASYNCcnt + workgroup barrier order the hand-off.
// ---------------------------------------------------------------------------
__global__ __launch_bounds__(128)
void vq_argmin_wmma_kernel(const float* __restrict__ z, const float* __restrict__ emb,
                           int* __restrict__ idx, int dbase, int Cz, int T, int N, int K)
{
    __shared__ __align__(16) float sbuf[2][1024];   // 2 x (16 codes x 64 dims)
    const int tid  = threadIdx.x;
    const int lane = threadIdx.x & 31;
    const int wave = blockIdx.x * (blockDim.x >> 5) + (threadIdx.x >> 5);
    // N/16 waves, launched as exactly (N/16)/4 blocks -> guard never splits a block
    if (wave >= (N >> 4)) return;
    const int m    = lane & 15;
    const int hi   = lane >> 4;
    const int koff = hi << 1;
    const int pt = wave * 16 + m;
    const int b  = pt / T;
    const int t  = pt - b * T;

    // Stage a 4KB code tile: 128 threads x 2 b128 async copies.
    auto stage = [&](int ct, int bufid) {
        const float* src = emb + (size_t)ct * 1024 + tid * 4;
        unsigned dst = (unsigned)(unsigned long long)&sbuf[bufid][tid * 4];
        async_copy_b128_to_lds(dst, src);
        async_copy_b128_to_lds(dst + 2048, src + 512);
    };

    stage(0, 0);                                    // overlap with A-fragment preload

    v2f a[16];
    #pragma unroll
    for (int s = 0; s < 16; ++s) {
        int d0 = 4 * s + koff;
        a[s].x = z[((size_t)b * Cz + dbase + d0)     * T + t];
        a[s].y = z[((size_t)b * Cz + dbase + d0 + 1) * T + t];
    }
    float best[8]; int bidx[8];
    #pragma unroll
    for (int v = 0; v < 8; ++v) { best[v] = 3.4e38f; bidx[v] = 0; }

    wait_asynccnt0();
    __syncthreads();

    const int nct = K >> 4;
    for (int ct = 0; ct < nct; ++ct) {
        const int cur = ct & 1;
        if (ct + 1 < nct) stage(ct + 1, cur ^ 1);   // prefetch next tile into LDS

        const int code = ct * 16 + m;
        const float* erow = &sbuf[cur][m * 64];     // LDS reads -> ds_load_b64
        v8f acc = {};
        float eep = 0.f;
        #pragma unroll
        for (int s = 0; s < 16; ++s) {
            v2f bf = *(const v2f*)(erow + 4 * s + koff);
            eep += bf.x * bf.x + bf.y * bf.y;
            acc = wmma_f32(a[s], bf, acc);
        }
        float ee = eep + __shfl_xor(eep, 16, 32);
        #pragma unroll
        for (int v = 0; v < 8; ++v) {
            float d = ee - 2.f * acc[v];
            if (d < best[v]) { best[v] = d; bidx[v] = code; }
        }
        wait_asynccnt0();                            // next tile resident in LDS
        __syncthreads();                             // publish to all 4 waves
    }
    #pragma unroll
    for (int v = 0; v < 8; ++v) {
        #pragma unroll
        for (int off = 1; off < 16; off <<= 1) {
            float od = __shfl_xor(best[v], off, 32);
            int   oi = __shfl_xor(bidx[v], off, 32);
            if (od < best[v] || (od == best[v] && oi < bidx[v])) { best[v] = od; bidx[v] = oi; }
        }
        if ((lane & 15) == 0) idx[wave * 16 + v + hi * 8] = bidx[v];
    }
}

__global__ void vq_gather_kernel(const float* __restrict__ z, const float* __restrict__ emb,
                                 const int* __restrict__ idx, float* __restrict__ q,
                                 float* __restrict__ counts, float* __restrict__ acc_sq,
                                 int dbase, int Cz, int T, int N)
{
    int tid = blockIdx.x * blockDim.x + threadIdx.x;
    float sq = 0.f;
    if (tid < N * 64) {
        int n = tid >> 6, d = tid & 63;
        int b = n / T, t = n - b * T;
        int code = idx[n];
        float e = emb[(size_t)code * 64 + d];
        size_t zoff = ((size_t)b * Cz + dbase + d) * T + t;
        float diff = e - z[zoff];
        sq = diff * diff;
        q[zoff] = e;
        if (d == 0) atomicAdd(&counts[code], 1.f);
    }
    __shared__ float red[256];
    red[threadIdx.x] = sq; __syncthreads();
    for (int s = 128; s > 0; s >>= 1) {
        if (threadIdx.x < s) red[threadIdx.x] += red[threadIdx.x + s];
        __syncthreads();
    }
    if (threadIdx.x == 0) atomicAdd(acc_sq, red[0]);
}

__global__ void vq_stats_kernel(const float* __restrict__ counts, int K, float invN,
                                float* __restrict__ perp_out, float* __restrict__ usage_out)
{
    __shared__ float r1[256], r2[256];
    float s1 = 0.f, s2 = 0.f;
    float fK = (float)K;
    for (int i = threadIdx.x; i < K; i += blockDim.x) {
        float avg = counts[i] * invN;
        s1 += avg * logf(avg + 1e-10f);
        s2 += avg * logf(avg * fK + 1e-10f);
    }
    r1[threadIdx.x] = s1; r2[threadIdx.x] = s2; __syncthreads();
    for (int s = 128; s > 0; s >>= 1) {
        if (threadIdx.x < s) { r1[threadIdx.x] += r1[threadIdx.x + s];
                               r2[threadIdx.x] += r2[threadIdx.x + s]; }
        __syncthreads();
    }
    if (threadIdx.x == 0) { perp_out[0] = expf(-r1[0]); atomicAdd(usage_out, r2[0]); }
}

__global__ void vq_init_kernel(float* c1, float* c2, float* acc_sq, float* usage_slot)
{
    int i = blockIdx.x * blockDim.x + threadIdx.x;
    if (i < 4096) { c1[i] = 0.f; c2[i] = 0.f; }
    if (i == 0) { acc_sq[0] = 0.f; usage_slot[0] = 0.f; }
}

__global__ void vq_commit_kernel(const float* __restrict__ acc_sq,
                                 float* __restrict__ commit_out, float scale)
{
    if (threadIdx.x == 0 && blockIdx.x == 0) commit_out[0] = scale * acc_sq[0];
}

// ---------------------------------------------------------------------------
extern "C" void kernel_launch(void* const* d_in, const int* in_sizes, int n_in,
                              void* d_out, int out_size, void* d_ws, size_t ws_size,
                              hipStream_t stream)
{
    (void)in_sizes; (void)n_in; (void)out_size; (void)ws_size;
    const float* x      = (const float*)d_in[0];
    const float* e_c1_w = (const float*)d_in[1];  const float* e_c1_b = (const float*)d_in[2];
    const float* e_r1_w1= (const float*)d_in[3];  const float* e_r1_b1= (const float*)d_in[4];
    const float* e_r1_w2= (const float*)d_in[5];  const float* e_r1_b2= (const float*)d_in[6];
    const float* e_c2_w = (const float*)d_in[7];  const float* e_c2_b = (const float*)d_in[8];
    const float* e_r2_w1= (const float*)d_in[9];  const float* e_r2_b1= (const float*)d_in[10];
    const float* e_r2_w2= (const float*)d_in[11]; const float* e_r2_b2= (const float*)d_in[12];
    const float* e_c3_w = (const float*)d_in[13]; const float* e_c3_b = (const float*)d_in[14];
    const float* e_r3_w1= (const float*)d_in[15]; const float* e_r3_b1= (const float*)d_in[16];
    const float* e_r3_w2= (const float*)d_in[17]; const float* e_r3_b2= (const float*)d_in[18];
    const float* e_pr_w = (const float*)d_in[19]; const float* e_pr_b = (const float*)d_in[20];
    const float* emb1   = (const float*)d_in[21]; const float* emb2   = (const float*)d_in[22];
    const float* d_pr_w = (const float*)d_in[23]; const float* d_pr_b = (const float*)d_in[24];
    const float* d_r1_w1= (const float*)d_in[25]; const float* d_r1_b1= (const float*)d_in[26];
    const float* d_r1_w2= (const float*)d_in[27]; const float* d_r1_b2= (const float*)d_in[28];
    const float* d_u1_w = (const float*)d_in[29]; const float* d_u1_b = (const float*)d_in[30];
    const float* d_r2_w1= (const float*)d_in[31]; const float* d_r2_b1= (const float*)d_in[32];
    const float* d_r2_w2= (const float*)d_in[33]; const float* d_r2_b2= (const float*)d_in[34];
    const float* d_u2_w = (const float*)d_in[35]; const float* d_u2_b = (const float*)d_in[36];
    const float* d_r3_w1= (const float*)d_in[37]; const float* d_r3_b1= (const float*)d_in[38];
    const float* d_r3_w2= (const float*)d_in[39]; const float* d_r3_b2= (const float*)d_in[40];
    const float* d_u3_w = (const float*)d_in[41]; const float* d_u3_b = (const float*)d_in[42];
    const float* d_out_w= (const float*)d_in[43]; const float* d_out_b= (const float*)d_in[44];
    float* out = (float*)d_out;

    float* A  = (float*)d_ws;              // 16,777,216 floats
    float* Bb = A  + 16777216;             //  8,388,608 floats
    float* Cc = Bb + 8388608;              //  8,388,608 floats
    float* counts1 = Cc + 8388608;
    float* counts2 = counts1 + 4096;
    float* acc_sq  = counts2 + 4096;
    int*   idx1 = (int*)(acc_sq + 16);
    int*   idx2 = idx1 + 16384;

    const int NB = 16;
    auto blk = [](int waves) { return (waves + 3) / 4; };
    auto wavesConv  = [&](int Cout, int Tout) { return NB * ((Cout + 15) / 16) * (Tout / 64); };
    auto wavesConvT = [&](int Cout, int Tout) { return NB * (Cout / 16) * 2 * (Tout / 128); };

    #define CONV(KW,S,P,R,A_,E, xin,w_,b_,add_,y_, Cin,Tin,Cout,Tout)                         \
        conv1d_wmma_kernel<KW,S,P,R,A_,E,4><<<blk(wavesConv(Cout,Tout)), 128, 0, stream>>>(    \
            xin, w_, b_, add_, y_, NB, Cin, Tin, Cout, Tout)
    #define CONVT(xin,w_,b_,y_, Cin,Tin,Cout)                                                 \
        convT1d_wmma_kernel<4><<<blk(wavesConvT(Cout, 2*(Tin))), 128, 0, stream>>>(            \
            xin, w_, b_, y_, NB, Cin, Tin, Cout)

    auto resblocks = [&](float* xin, float* tmp, float* xout,
                         const float* w1, const float* b1,
                         const float* w2, const float* b2, int C, int T) {
        CONV(3,1,1,true,false,false,  xin, w1, b1, nullptr, tmp,  C, T, C, T);
        CONV(1,1,0,true,true, false,  tmp, w2, b2, xin,     xout, C, T, C, T);
        CONV(3,1,1,true,false,false,  xout, w1 + (size_t)C*C*3, b1 + C, nullptr, tmp, C, T, C, T);
        CONV(1,1,0,true,true, false,  tmp,  w2 + (size_t)C*C*1, b2 + C, xout,    xin, C, T, C, T);
    };

    // ---------------- Encoder ----------------
    CONV(7,2,3,false,false,true,  x, e_c1_w, e_c1_b, nullptr, Bb, 1, 8192, 128, 4096);
    resblocks(Bb, Cc, A, e_r1_w1, e_r1_b1, e_r1_w2, e_r1_b2, 128, 4096);          // -> Bb
    CONV(5,2,2,false,false,false, Bb, e_c2_w, e_c2_b, nullptr, A, 128, 4096, 256, 2048);
    resblocks(A, Cc, Bb, e_r2_w1, e_r2_b1, e_r2_w2, e_r2_b2, 256, 2048);          // -> A
    CONV(3,2,1,false,false,false, A, e_c3_w, e_c3_b, nullptr, Bb, 256, 2048, 384, 1024);
    resblocks(Bb, Cc, A, e_r3_w1, e_r3_b1, e_r3_w2, e_r3_b2, 384, 1024);          // -> Bb
    CONV(1,1,0,false,false,false, Bb, e_pr_w, e_pr_b, nullptr, A, 384, 1024, 128, 1024);

    // ---------------- Product VQ (z in A, q in Cc) ----------------
    const int N = 16 * 1024;
    vq_init_kernel<<<16, 256, 0, stream>>>(counts1, counts2, acc_sq, out + 131073);
    vq_argmin_wmma_kernel<<<(N/16)/4, 128, 0, stream>>>(A, emb1, idx1, 0,  128, 1024, N, 4096);
    vq_argmin_wmma_kernel<<<(N/16)/4, 128, 0, stream>>>(A, emb2, idx2, 64, 128, 1024, N, 4096);
    vq_gather_kernel<<<(N*64)/256, 256, 0, stream>>>(A, emb1, idx1, Cc, counts1, acc_sq, 0,  128, 1024, N);
    vq_gather_kernel<<<(N*64)/256, 256, 0, stream>>>(A, emb2, idx2, Cc, counts2, acc_sq, 64, 128, 1024, N);
    vq_stats_kernel<<<1, 256, 0, stream>>>(counts1, 4096, 1.f/(float)N, out + 131074, out + 131073);
    vq_stats_kernel<<<1, 256, 0, stream>>>(counts2, 4096, 1.f/(float)N, out + 131075, out + 131073);
    vq_commit_kernel<<<1, 1, 0, stream>>>(acc_sq, out + 131072, 0.25f / ((float)N * 64.f));

    // ---------------- Decoder ----------------
    CONV(1,1,0,false,false,false, Cc, d_pr_w, d_pr_b, nullptr, Bb, 128, 1024, 384, 1024);
    resblocks(Bb, A, Cc, d_r1_w1, d_r1_b1, d_r1_w2, d_r1_b2, 384, 1024);          // -> Bb
    CONVT(Bb, d_u1_w, d_u1_b, A, 384, 1024, 256);                                 // A [16,256,2048]
    resblocks(A, Cc, Bb, d_r2_w1, d_r2_b1, d_r2_w2, d_r2_b2, 256, 2048);          // -> A
    CONVT(A, d_u2_w, d_u2_b, Bb, 256, 2048, 128);                                 // Bb [16,128,4096]
    resblocks(Bb, Cc, A, d_r3_w1, d_r3_b1, d_r3_w2, d_r3_b2, 128, 4096);          // -> Bb
    CONVT(Bb, d_u3_w, d_u3_b, A, 128, 4096, 128);                                 // A [16,128,8192]
    CONV(7,1,3,false,false,true,  A, d_out_w, d_out_b, nullptr, out, 128, 8192, 1, 8192);

    #undef CONV
    #undef CONVT
}